// MambaBlock_8083128451419
// MI455X (gfx1250) — compile-verified
//
#include <hip/hip_runtime.h>
#include <math.h>

// MambaBlock on gfx1250 (CDNA5, wave32).
//  - GEMMs: V_WMMA_F32_16X16X4_F32 (fp32 precision; GEMM flops far below the
//    matrix-pipe ceiling, weights stay L2-resident at 192MB).
//  - Tile staging: Tensor Data Mover (tensor_load_to_lds) with descriptor
//    padding (2 dwords per 256) -> 516-float row stride, bank-conflict-free
//    and 8B-aligned for ds_load_b64 A-operands.
//  - Scan: wave32-native, register-shuffle broadcast of B_t/C_t.

typedef __attribute__((ext_vector_type(2))) float    v2f;
typedef __attribute__((ext_vector_type(8))) float    v8f;
typedef __attribute__((ext_vector_type(4))) unsigned v4u;
typedef __attribute__((ext_vector_type(8))) unsigned v8u;

#define DDIM 512
#define NDIM 16
#define LSEQ 1024
#define BSZ  8
#define RTOT (BSZ * LSEQ)          // 8192 token rows
#define LROWF 516                  // effective padded row stride (floats)

// TDM pad layout: 2 pad dwords inserted after every 256 dwords streamed.
__device__ __forceinline__ int pidx(int row, int col) {
  return row * LROWF + col + ((col >> 8) << 1);
}

__device__ __forceinline__ unsigned lds_off(const void* p) {
  return (unsigned)(unsigned long long)
         (__attribute__((address_space(3))) const void*)p;
}

// Issue one TDM DMA: 16 rows x 512 f32 from global (row stride 512) into LDS
// at byte offset lds_byte, with pad_interval=256dw / pad_amount=2dw.
__device__ __forceinline__ void tdm_load_16x512(const float* gsrc,
                                                unsigned lds_byte) {
  unsigned long long ga = (unsigned long long)gsrc;
  v4u g0;
  g0[0] = 1u;                                    // count=1 (valid), user mode
  g0[1] = lds_byte;                              // lds_addr
  g0[2] = (unsigned)ga;                          // global_addr[31:0]
  g0[3] = (unsigned)((ga >> 32) & 0x01FFFFFFull) // global_addr[56:32]
        | (2u << 30);                            // type=2 ("image")
  v8u g1;
  g1[0] = (2u << 16)                             // data_size = 4B
        | (1u << 20)                             // pad_enable
        | (7u << 22)                             // pad_interval: 256 dwords
        | (1u << 25);                            // pad_amount: 2 dwords
  g1[1] = (512u << 16);                          // tensor_dim0 = 512 (lo16)
  g1[2] = (16u  << 16);                          // tensor_dim1 = 16  (lo16)
  g1[3] = (512u << 16);                          // tile_dim0 = 512
  g1[4] = 16u;                                   // tile_dim1 = 16, tile_dim2=0
  g1[5] = 512u;                                  // tensor_dim0_stride lo32
  g1[6] = 0u;
  g1[7] = 0u;
  asm volatile("tensor_load_to_lds %0, %1" :: "s"(g0), "s"(g1) : "memory");
}

__device__ __forceinline__ v8f wmma_k4(v2f a, v2f b, v8f c) {
  return __builtin_amdgcn_wmma_f32_16x16x4_f32(false, a, false, b, (short)0, c,
                                               false, false);
}

// ---------------------------------------------------------------------------
// Kernel 1: u = x*Wi^T + bi ; z = x*Wz^T + bz ; Bp = u*WB^T + bB ; Cp = u*WC^T + bC
// ---------------------------------------------------------------------------
__global__ __launch_bounds__(128)
void mamba_proj_kernel(const float* __restrict__ x,
                       const float* __restrict__ Wi, const float* __restrict__ bi,
                       const float* __restrict__ Wz, const float* __restrict__ bz,
                       const float* __restrict__ WB, const float* __restrict__ bB,
                       const float* __restrict__ WC, const float* __restrict__ bC,
                       float* __restrict__ u, float* __restrict__ z,
                       float* __restrict__ Bp, float* __restrict__ Cp)
{
  __shared__ float smem[2 * 16 * LROWF];
  float* xs = smem;                 // x tile (TDM-filled, padded layout)
  float* us = smem + 16 * LROWF;    // u tile (WMMA-filled, same layout)

  const int tid  = threadIdx.x;
  const int wave = tid >> 5;
  const int lane = tid & 31;
  const int half = lane >> 4;       // K sub-block select (K=0,1 vs K=2,3)
  const int lid  = lane & 15;
  const int r0   = blockIdx.x * 16;

  if (tid < 32) {                   // wave 0 drives the tile DMA
    tdm_load_16x512(x + (size_t)r0 * DDIM, lds_off(xs));
    __builtin_amdgcn_s_wait_tensorcnt(0);
  }
  __syncthreads();

  const bool   is_u = (wave < 2);
  const float* W    = is_u ? Wi : Wz;
  const float* bias = is_u ? bi : bz;
  float*       outp = is_u ? u  : z;
  const int tbase = (wave & 1) * 16;     // 16 column-tiles per wave, 8 pairs

  for (int p = 0; p < 8; ++p) {
    const int n0 = (tbase + 2 * p) * 16;
    v8f acc0 = {}; v8f acc1 = {};
    const float* w0 = W + (size_t)(n0 + lid) * DDIM + 2 * half;
    const float* w1 = w0 + 16 * DDIM;
    #pragma unroll 4
    for (int k = 0; k < DDIM; k += 4) {
      v2f a  = *(const v2f*)&xs[pidx(lid, k + 2 * half)];  // ds_load_b64
      v2f b0 = *(const v2f*)(w0 + k);                      // global_load_b64
      v2f b1 = *(const v2f*)(w1 + k);
      acc0 = wmma_k4(a, b0, acc0);
      acc1 = wmma_k4(a, b1, acc1);
    }
    const float bn0 = bias[n0 + lid];
    const float bn1 = bias[n0 + 16 + lid];
    #pragma unroll
    for (int j = 0; j < 8; ++j) {
      const int row = j + 8 * half;                        // C/D: M = j + 8*half
      float v0 = acc0[j] + bn0;
      float v1 = acc1[j] + bn1;
      outp[(size_t)(r0 + row) * DDIM + n0 + lid]      = v0;
      outp[(size_t)(r0 + row) * DDIM + n0 + 16 + lid] = v1;
      if (is_u) {
        us[pidx(row, n0 + lid)]      = v0;
        us[pidx(row, n0 + 16 + lid)] = v1;
      }
    }
  }
  __syncthreads();

  // Bp (wave 0) / Cp (wave 1): one 16x16 tile each, K=512 from LDS u tile.
  if (wave < 2) {
    const float* Wn = wave ? WC : WB;
    const float* bn = wave ? bC : bB;
    float*       op = wave ? Cp : Bp;
    v8f acc = {};
    const float* w0 = Wn + (size_t)lid * DDIM + 2 * half;
    #pragma unroll 4
    for (int k = 0; k < DDIM; k += 4) {
      v2f a = *(const v2f*)&us[pidx(lid, k + 2 * half)];
      v2f b = *(const v2f*)(w0 + k);
      acc = wmma_k4(a, b, acc);
    }
    const float bb = bn[lid];
    #pragma unroll
    for (int j = 0; j < 8; ++j) {
      const int row = j + 8 * half;
      op[(size_t)(r0 + row) * NDIM + lid] = acc[j] + bb;
    }
  }
}

// ---------------------------------------------------------------------------
// Kernel 2: selective scan + D-skip + silu(z) gate. One wave per
// (b, 32-channel slice); B_t/C_t broadcast by __shfl, y in-place over z.
// ---------------------------------------------------------------------------
__global__ __launch_bounds__(32)
void mamba_scan_kernel(const float* __restrict__ u, float* __restrict__ zy,
                       const float* __restrict__ Bp, const float* __restrict__ Cp,
                       const float* __restrict__ A,  const float* __restrict__ Dv,
                       const float* __restrict__ delta_bias)
{
  const int lane = threadIdx.x;
  const int b    = blockIdx.x >> 4;
  const int d    = ((blockIdx.x & 15) << 5) + lane;

  float a[NDIM];
  #pragma unroll
  for (int n = 0; n < NDIM; ++n) a[n] = A[d * NDIM + n];
  const float dbias = delta_bias[d];
  const float Dvd   = Dv[d];

  float h[NDIM];
  #pragma unroll
  for (int n = 0; n < NDIM; ++n) h[n] = 0.f;

  const float* bcp = (lane < 16) ? (Bp + lane) : (Cp + (lane - 16));
  const size_t rbase = (size_t)b * LSEQ;

  for (int l = 0; l < LSEQ; ++l) {
    const size_t r = rbase + l;
    const float bc = bcp[r * NDIM];
    const float ut = u [r * DDIM + d];
    const float zt = zy[r * DDIM + d];

    const float sx = ut + dbias;
    const float dt = (sx > 20.f) ? sx : log1pf(__expf(sx));   // softplus
    const float du = dt * ut;

    float yv = 0.f;
    #pragma unroll
    for (int n = 0; n < NDIM; ++n) {
      const float Bn = __shfl(bc, n, 32);
      const float Cn = __shfl(bc, n + 16, 32);
      h[n] = __expf(dt * a[n]) * h[n] + du * Bn;
      yv  += h[n] * Cn;
    }
    yv += ut * Dvd;
    const float gate = zt / (1.f + __expf(-zt));              // silu(z)
    zy[r * DDIM + d] = yv * gate;
  }
}

// ---------------------------------------------------------------------------
// Kernel 3: yo = y*Wo^T + bo ; out = x + LayerNorm(yo)*gamma + beta
// ---------------------------------------------------------------------------
__global__ __launch_bounds__(128)
void mamba_outln_kernel(const float* __restrict__ y,  const float* __restrict__ Wo,
                        const float* __restrict__ bo, const float* __restrict__ gamma,
                        const float* __restrict__ beta, const float* __restrict__ x,
                        float* __restrict__ out)
{
  __shared__ float smem[2 * 16 * LROWF];
  float* ys = smem;
  float* os = smem + 16 * LROWF;

  const int tid  = threadIdx.x;
  const int wave = tid >> 5;
  const int lane = tid & 31;
  const int half = lane >> 4;
  const int lid  = lane & 15;
  const int r0   = blockIdx.x * 16;

  if (tid < 32) {
    tdm_load_16x512(y + (size_t)r0 * DDIM, lds_off(ys));
    __builtin_amdgcn_s_wait_tensorcnt(0);
  }
  __syncthreads();

  for (int p = 0; p < 4; ++p) {
    const int n0 = (wave * 8 + 2 * p) * 16;
    v8f acc0 = {}; v8f acc1 = {};
    const float* w0 = Wo + (size_t)(n0 + lid) * DDIM + 2 * half;
    const float* w1 = w0 + 16 * DDIM;
    #pragma unroll 4
    for (int k = 0; k < DDIM; k += 4) {
      v2f a = *(const v2f*)&ys[pidx(lid, k + 2 * half)];
      acc0 = wmma_k4(a, *(const v2f*)(w0 + k), acc0);
      acc1 = wmma_k4(a, *(const v2f*)(w1 + k), acc1);
    }
    const float bn0 = bo[n0 + lid];
    const float bn1 = bo[n0 + 16 + lid];
    #pragma unroll
    for (int j = 0; j < 8; ++j) {
      const int row = j + 8 * half;
      os[pidx(row, n0 + lid)]      = acc0[j] + bn0;
      os[pidx(row, n0 + 16 + lid)] = acc1[j] + bn1;
    }
  }
  __syncthreads();

  // LayerNorm + residual: each wave owns 4 rows; butterfly reduce in-wave.
  for (int rr = 0; rr < 4; ++rr) {
    const int row = wave * 4 + rr;
    float s1 = 0.f, s2 = 0.f;
    #pragma unroll
    for (int j = 0; j < 16; ++j) {
      const float v = os[pidx(row, lane + 32 * j)];
      s1 += v; s2 += v * v;
    }
    #pragma unroll
    for (int off = 16; off > 0; off >>= 1) {
      s1 += __shfl_xor(s1, off, 32);
      s2 += __shfl_xor(s2, off, 32);
    }
    const float mu   = s1 * (1.f / 512.f);
    const float var  = s2 * (1.f / 512.f) - mu * mu;
    const float rstd = rsqrtf(var + 1e-5f);
    #pragma unroll
    for (int j = 0; j < 16; ++j) {
      const int col = lane + 32 * j;
      const float ln = (os[pidx(row, col)] - mu) * rstd * gamma[col] + beta[col];
      const size_t gi = (size_t)(r0 + row) * DDIM + col;
      out[gi] = x[gi] + ln;
    }
  }
}

// ---------------------------------------------------------------------------
extern "C" void kernel_launch(void* const* d_in, const int* in_sizes, int n_in,
                              void* d_out, int out_size, void* d_ws, size_t ws_size,
                              hipStream_t stream) {
  (void)in_sizes; (void)n_in; (void)out_size; (void)ws_size;
  const float* x     = (const float*)d_in[0];
  const float* Wi    = (const float*)d_in[1];
  const float* bi    = (const float*)d_in[2];
  const float* Wz    = (const float*)d_in[3];
  const float* bz    = (const float*)d_in[4];
  const float* A     = (const float*)d_in[5];
  const float* Dv    = (const float*)d_in[6];
  const float* dbias = (const float*)d_in[7];
  const float* WB    = (const float*)d_in[8];
  const float* bB    = (const float*)d_in[9];
  const float* WC    = (const float*)d_in[10];
  const float* bC    = (const float*)d_in[11];
  const float* Wo    = (const float*)d_in[12];
  const float* bo    = (const float*)d_in[13];
  const float* gamma = (const float*)d_in[14];
  const float* beta  = (const float*)d_in[15];
  float* out = (float*)d_out;

  float* ws = (float*)d_ws;
  float* u  = ws;                                   // RTOT*DDIM
  float* zy = u  + (size_t)RTOT * DDIM;             // RTOT*DDIM (z, then y)
  float* Bp = zy + (size_t)RTOT * DDIM;             // RTOT*NDIM
  float* Cp = Bp + (size_t)RTOT * NDIM;             // RTOT*NDIM

  mamba_proj_kernel<<<RTOT / 16, 128, 0, stream>>>(x, Wi, bi, Wz, bz, WB, bB,
                                                   WC, bC, u, zy, Bp, Cp);
  mamba_scan_kernel<<<BSZ * (DDIM / 32), 32, 0, stream>>>(u, zy, Bp, Cp, A, Dv,
                                                          dbias);
  mamba_outln_kernel<<<RTOT / 16, 128, 0, stream>>>(zy, Wo, bo, gamma, beta, x,
                                                    out);
}